// NearestMean_20667382629126
// MI455X (gfx1250) — compile-verified
//
#include <hip/hip_runtime.h>

// NearestMean bucketization: out[i] = labels[searchsorted(thresholds, X[i], side='right')]
// X: 142,848,000 float32 (16x240x240x155). thresholds: 3 float32. labels: 4 int32.
// Pure streaming kernel: 1.14 GB of traffic -> ~49us floor at 23.3 TB/s HBM.
// Design: 128-bit non-temporal loads/stores (stream-once data must not pollute
// the 192MB L2), branchless v_cmp/v_cndmask classify, grid-stride wave32 blocks.

typedef float vf4 __attribute__((ext_vector_type(4)));
typedef int   vi4 __attribute__((ext_vector_type(4)));
typedef long long i64;

__device__ __forceinline__ int classify1(float x, float t0, float t1, float t2,
                                         int l0, int l1, int l2, int l3) {
    // searchsorted side='right': x == t goes to the NEXT bucket, i.e. strict '<' picks this bucket.
    // Lowers to 3x v_cmp_lt_f32 + 3x v_cndmask_b32 (branchless).
    int r = l3;
    r = (x < t2) ? l2 : r;
    r = (x < t1) ? l1 : r;
    r = (x < t0) ? l0 : r;
    return r;
}

__global__ __launch_bounds__(256) void nearest_mean_kernel(
    const float* __restrict__ X,
    const float* __restrict__ thr,
    const int*   __restrict__ labs,
    int*         __restrict__ out,
    i64 n)
{
    // Uniform (kernel-arg-relative) loads -> scalar s_load, stays in SGPRs for the whole loop.
    const float t0 = thr[0], t1 = thr[1], t2 = thr[2];
    const int   l0 = labs[0], l1 = labs[1], l2 = labs[2], l3 = labs[3];

    const i64 nvec = n >> 2;                       // whole-vec4 portion
    const vf4* __restrict__ X4 = (const vf4*)X;
    vi4*       __restrict__ O4 = (vi4*)out;

    const i64 tid    = (i64)blockIdx.x * blockDim.x + threadIdx.x;
    const i64 stride = (i64)gridDim.x * blockDim.x;

    for (i64 i = tid; i < nvec; i += stride) {
        // Non-temporal 128-bit load: global_load_b128 with NT temporal hint.
        vf4 v = __builtin_nontemporal_load(&X4[i]);
        vi4 r;
        r.x = classify1(v.x, t0, t1, t2, l0, l1, l2, l3);
        r.y = classify1(v.y, t0, t1, t2, l0, l1, l2, l3);
        r.z = classify1(v.z, t0, t1, t2, l0, l1, l2, l3);
        r.w = classify1(v.w, t0, t1, t2, l0, l1, l2, l3);
        // Non-temporal 128-bit store: global_store_b128, NT (stream-once output).
        __builtin_nontemporal_store(r, &O4[i]);
    }

    // Tail (n % 4 elements) — at most 3 scalar elements, handled by the first threads.
    const i64 base = nvec << 2;
    const i64 rem  = n - base;
    if (tid < rem) {
        float x = X[base + tid];
        out[base + tid] = classify1(x, t0, t1, t2, l0, l1, l2, l3);
    }
}

extern "C" void kernel_launch(void* const* d_in, const int* in_sizes, int n_in,
                              void* d_out, int out_size, void* d_ws, size_t ws_size,
                              hipStream_t stream) {
    const float* X    = (const float*)d_in[0];   // float32 volume
    const float* thr  = (const float*)d_in[1];   // 3 thresholds
    const int*   labs = (const int*)d_in[2];     // 4 labels (int32)
    int*         out  = (int*)d_out;             // int32 output, same shape as X

    const i64 n     = (i64)in_sizes[0];          // 142,848,000
    const int block = 256;                       // 8 wave32s per block
    i64 nvec   = (n + 3) >> 2;
    i64 blocks = (nvec + block - 1) / block;     // ~139.5K blocks: one vec4/thread,
    if (blocks < 1) blocks = 1;                  // maximal memory-level parallelism
    if (blocks > 0x7FFFFFFFLL) blocks = 0x7FFFFFFFLL;

    nearest_mean_kernel<<<dim3((unsigned)blocks), dim3(block), 0, stream>>>(
        X, thr, labs, out, n);
}